// Encoder2_77618648973416
// MI455X (gfx1250) — compile-verified
//
#include <hip/hip_runtime.h>

#define N_NODES 100000
#define D 128
#define N_EDGES 1600000
#define BN_EPS 1e-5f

typedef float v2f __attribute__((ext_vector_type(2)));
typedef float v8f __attribute__((ext_vector_type(8)));

// ---------------------------------------------------------------------------
// Kernel 1: X = heat @ W  (f32 GEMM via V_WMMA_F32_16X16X4_F32, wave32)
// Block = 256 threads = 8 waves. Block handles a 16-row strip of heat; wave w
// produces the 16x16 output tile at columns [16w, 16w+16).
// ---------------------------------------------------------------------------
__global__ __launch_bounds__(256) void gemm_wmma_f32(const float* __restrict__ A,
                                                     const float* __restrict__ W,
                                                     float* __restrict__ X) {
    __shared__ float As[16][132];  // 16x128 tile, padded to avoid bank conflicts

    const int t  = threadIdx.x;
    const int m0 = blockIdx.x * 16;

    // Cooperative load of the 16x128 A tile (2048 floats / 256 threads).
#pragma unroll
    for (int i = 0; i < 8; ++i) {
        int idx = t + i * 256;
        int r = idx >> 7;
        int c = idx & 127;
        As[r][c] = A[(size_t)(m0 + r) * D + c];
    }
    __syncthreads();

    const int lane = t & 31;
    const int n0   = (t >> 5) * 16;     // wave's output column base
    const int half = lane >> 4;         // 0: holds K+0/K+1 ; 1: holds K+2/K+3
    const int mr   = lane & 15;         // row (for A) / col (for B) within tile

    v8f c = {};
#pragma unroll 4
    for (int k0 = 0; k0 < D; k0 += 4) {
        const int ka = k0 + 2 * half;
        v2f a, b;
        // A 16x4 f32 layout: VGPR0 = K (half?2:0), VGPR1 = K (half?3:1), lanes%16 = M
        a.x = As[mr][ka + 0];
        a.y = As[mr][ka + 1];
        // B 4x16 f32 layout mirrors A: rows striped across lanes within a VGPR
        b.x = W[(size_t)(ka + 0) * D + n0 + mr];
        b.y = W[(size_t)(ka + 1) * D + n0 + mr];
        c = __builtin_amdgcn_wmma_f32_16x16x4_f32(false, a, false, b,
                                                  (short)0, c, false, false);
    }

    // C/D layout: VGPR v -> row m0 + v + 8*half, col n0 + mr
#pragma unroll
    for (int v = 0; v < 8; ++v) {
        X[(size_t)(m0 + v + 8 * half) * D + n0 + mr] = c[v];
    }
}

// ---------------------------------------------------------------------------
// Kernel 2: edge gather/scale/scatter-add. One wave per edge; lane l moves
// float4 at feature offset 4l. x rows + agg rows are L2-resident (51 MB each,
// 192 MB L2), atomics resolved at the L2 atomic units.
// ---------------------------------------------------------------------------
__global__ __launch_bounds__(256) void edge_scatter(const float* __restrict__ x,
                                                    const int* __restrict__ src,
                                                    const int* __restrict__ dst,
                                                    const float* __restrict__ ew,
                                                    float* __restrict__ agg) {
    int gid  = blockIdx.x * 256 + threadIdx.x;
    int e    = gid >> 5;
    int lane = gid & 31;
    if (e >= N_EDGES) return;

    int   s = src[e];             // wave-uniform (all lanes same address -> L0 hit)
    int   d = dst[e];
    float w = ew[e];

    const float4* xr = (const float4*)(x + (size_t)s * D);
    float4 v = xr[lane];
    v.x *= w; v.y *= w; v.z *= w; v.w *= w;

    float* ar = agg + (size_t)d * D + lane * 4;
    unsafeAtomicAdd(ar + 0, v.x);   // global_atomic_add_f32 (no CAS loop)
    unsafeAtomicAdd(ar + 1, v.y);
    unsafeAtomicAdd(ar + 2, v.z);
    unsafeAtomicAdd(ar + 3, v.w);
}

// ---------------------------------------------------------------------------
// Kernel 3: h = prelu(agg + b, a1) in place; accumulate per-feature sum and
// sum-of-squares. 128 threads/block (thread == feature), grid-stride over
// rows, one pair of atomics per (block, feature) at the end.
// ---------------------------------------------------------------------------
__global__ __launch_bounds__(128) void bias_act_stats(float* __restrict__ agg,
                                                      const float* __restrict__ b,
                                                      const float* __restrict__ a1,
                                                      float* __restrict__ sums,
                                                      float* __restrict__ sumsq) {
    const int   f     = threadIdx.x;
    const float bias  = b[f];
    const float alpha = a1[0];

    float s = 0.0f, sq = 0.0f;
    for (int r = blockIdx.x; r < N_NODES; r += gridDim.x) {
        size_t idx = (size_t)r * D + f;
        float v = agg[idx] + bias;
        v = (v >= 0.0f) ? v : alpha * v;
        agg[idx] = v;
        s  += v;
        sq += v * v;
    }
    unsafeAtomicAdd(&sums[f],  s);
    unsafeAtomicAdd(&sumsq[f], sq);
}

// ---------------------------------------------------------------------------
// Kernel 4a: fold BN statistics into per-feature scale/shift. 1 block.
// ---------------------------------------------------------------------------
__global__ __launch_bounds__(128) void bn_params(const float* __restrict__ sums,
                                                 const float* __restrict__ sumsq,
                                                 const float* __restrict__ gamma,
                                                 const float* __restrict__ beta,
                                                 float* __restrict__ scale,
                                                 float* __restrict__ shift) {
    const int f = threadIdx.x;
    const float invN = 1.0f / (float)N_NODES;
    float mean = sums[f] * invN;
    float var  = sumsq[f] * invN - mean * mean;   // biased variance
    float sc   = gamma[f] * rsqrtf(var + BN_EPS);
    scale[f] = sc;
    shift[f] = beta[f] - mean * sc;
}

// ---------------------------------------------------------------------------
// Kernel 4b: out = prelu(h * scale + shift, a2), float4 vectorized.
// ---------------------------------------------------------------------------
__global__ __launch_bounds__(256) void bn_prelu_out(const float* __restrict__ h,
                                                    const float* __restrict__ scale,
                                                    const float* __restrict__ shift,
                                                    const float* __restrict__ a2,
                                                    float* __restrict__ out) {
    int t = blockIdx.x * 256 + threadIdx.x;
    const int n4 = (N_NODES * D) / 4;
    if (t >= n4) return;

    const float alpha = a2[0];
    int f4 = t & 31;  // 128 features / 4 per float4

    float4 v  = ((const float4*)h)[t];
    float4 sc = ((const float4*)scale)[f4];
    float4 sh = ((const float4*)shift)[f4];

    float4 r;
    r.x = fmaf(v.x, sc.x, sh.x);
    r.y = fmaf(v.y, sc.y, sh.y);
    r.z = fmaf(v.z, sc.z, sh.z);
    r.w = fmaf(v.w, sc.w, sh.w);
    r.x = (r.x >= 0.0f) ? r.x : alpha * r.x;
    r.y = (r.y >= 0.0f) ? r.y : alpha * r.y;
    r.z = (r.z >= 0.0f) ? r.z : alpha * r.z;
    r.w = (r.w >= 0.0f) ? r.w : alpha * r.w;

    ((float4*)out)[t] = r;
}

// ---------------------------------------------------------------------------
// Workspace layout (floats):
//   [0, N*D)           agg / h
//   [N*D, N*D+128)     per-feature sums
//   [.., +128)         per-feature sumsq
//   [.., +128)         BN scale
//   [.., +128)         BN shift
// x (= heat @ W) is staged in d_out and consumed by the edge kernel before
// the final pass overwrites d_out.
// ---------------------------------------------------------------------------
extern "C" void kernel_launch(void* const* d_in, const int* in_sizes, int n_in,
                              void* d_out, int out_size, void* d_ws, size_t ws_size,
                              hipStream_t stream) {
    const float* heat  = (const float*)d_in[0];
    const int*   src   = (const int*)  d_in[1];
    const int*   dst   = (const int*)  d_in[2];
    const float* ew    = (const float*)d_in[3];
    const float* W     = (const float*)d_in[4];
    const float* b     = (const float*)d_in[5];
    const float* a1    = (const float*)d_in[6];
    const float* gamma = (const float*)d_in[7];
    const float* beta  = (const float*)d_in[8];
    const float* a2    = (const float*)d_in[9];
    float*       out   = (float*)d_out;

    float* agg   = (float*)d_ws;
    float* sums  = agg + (size_t)N_NODES * D;
    float* sumsq = sums + D;
    float* scale = sumsq + D;
    float* shift = scale + D;

    // Zero agg + sums + sumsq (graph-capturable async memset).
    hipMemsetAsync(agg, 0, ((size_t)N_NODES * D + 2 * D) * sizeof(float), stream);

    // 1) x = heat @ W  -> d_out   (6250 blocks * 8 waves: one 16x16 tile/wave)
    gemm_wmma_f32<<<N_NODES / 16, 256, 0, stream>>>(heat, W, out);

    // 2) scatter-add over edges (one wave per edge)
    edge_scatter<<<(N_EDGES * 32) / 256, 256, 0, stream>>>(out, src, dst, ew, agg);

    // 3) bias + PReLU + batch stats
    bias_act_stats<<<1024, 128, 0, stream>>>(agg, b, a1, sums, sumsq);

    // 4a) BN scale/shift
    bn_params<<<1, 128, 0, stream>>>(sums, sumsq, gamma, beta, scale, shift);

    // 4b) BN apply + outer PReLU -> d_out
    bn_prelu_out<<<((N_NODES * D) / 4 + 255) / 256, 256, 0, stream>>>(
        agg, scale, shift, a2, out);
}